// TwoConv_14242111553626
// MI455X (gfx1250) — compile-verified
//
#include <hip/hip_runtime.h>
#include <hip/hip_bf16.h>

typedef __attribute__((ext_vector_type(2))) float v2f;
typedef __attribute__((ext_vector_type(8))) float v8f;

#define N_NODES 100000
#define HEADS   4

// ---------------------------------------------------------------------------
// Zero-fill scratch (graph-capture-safe; no hipMemsetAsync dependence)
// ---------------------------------------------------------------------------
__global__ void zero_kernel(float* __restrict__ p, size_t n) {
    size_t i = (size_t)blockIdx.x * blockDim.x + threadIdx.x;
    if (i < n) p[i] = 0.0f;
}

// ---------------------------------------------------------------------------
// Node GEMM: T[N,OUT] = X[N,16] @ W[16,OUT]  via V_WMMA_F32_16X16X4_F32
//            D[N,4]   = X[N,16] @ U[16,4]    via VALU (4 cols, too narrow for WMMA)
// One wave32 per 16-node tile. K=16 split into 4 chained K=4 WMMAs.
// fp32 WMMA => bit-compatible precision with the fp32 reference math.
//
// A 16x4 f32 layout (ISA 7.12.2): lanes 0-15 M=0..15; VGPR v: K=(lane/16)*2+v
// B 4x16 layout mirrors A (row striped across lanes):  K=(lane/16)*2+v, N=lane%16
// C/D 16x16 f32: VGPR v -> M = v + (lane/16)*8, N = lane%16
// ---------------------------------------------------------------------------
template <int OUT>
__global__ void feast_node_gemm(const float* __restrict__ X,   // [n,16]
                                const float* __restrict__ W,   // [16,OUT]
                                const float* __restrict__ U,   // [16,4]
                                float* __restrict__ T,         // [n,OUT]
                                float* __restrict__ D,         // [n,4]
                                int n) {
    const int lane  = threadIdx.x & 31;
    const int wave  = threadIdx.x >> 5;
    const int tile  = blockIdx.x * (blockDim.x >> 5) + wave;
    const int node0 = tile * 16;
    if (node0 >= n) return;                 // uniform per wave

    const int row = lane & 15;              // M index / N index
    const int kh  = lane >> 4;              // 0 or 1 (K half)

    const int arow = min(node0 + row, n - 1);   // clamp loads; guard stores below

    // ---- A fragments: 4 chunks of K=4, each lane holds 2 contiguous floats
    v2f a[4];
    const float* xr = X + (size_t)arow * 16 + kh * 2;
#pragma unroll
    for (int kk = 0; kk < 4; ++kk)
        a[kk] = *(const v2f*)(xr + kk * 4);

    // ---- WMMA over OUT/16 column tiles
    constexpr int NT = OUT / 16;
    v8f c[NT];
#pragma unroll
    for (int nt = 0; nt < NT; ++nt) {
        v8f accv = {};
#pragma unroll
        for (int kk = 0; kk < 4; ++kk) {
            v2f b;
            b.x = W[(size_t)(kk * 4 + kh * 2 + 0) * OUT + nt * 16 + row];
            b.y = W[(size_t)(kk * 4 + kh * 2 + 1) * OUT + nt * 16 + row];
            accv = __builtin_amdgcn_wmma_f32_16x16x4_f32(
                false, a[kk], false, b, (short)0, accv, false, false);
        }
        c[nt] = accv;
    }

    // ---- store D-matrix: M = v + kh*8, N = row
#pragma unroll
    for (int nt = 0; nt < NT; ++nt) {
#pragma unroll
        for (int v = 0; v < 8; ++v) {
            int m = node0 + v + kh * 8;
            if (m < n) T[(size_t)m * OUT + nt * 16 + row] = c[nt][v];
        }
    }

    // ---- d = X @ U : each lane computes 2 of the 4 head dots for its row
    if (node0 + row < n) {
        const float* xn = X + (size_t)(node0 + row) * 16;
        float d0 = 0.0f, d1 = 0.0f;
#pragma unroll
        for (int k = 0; k < 16; ++k) {
            float xv = xn[k];
            d0 += xv * U[k * 4 + kh * 2 + 0];
            d1 += xv * U[k * 4 + kh * 2 + 1];
        }
        D[(size_t)(node0 + row) * 4 + kh * 2 + 0] = d0;
        D[(size_t)(node0 + row) * 4 + kh * 2 + 1] = d1;
    }
}

// ---------------------------------------------------------------------------
// Edge kernel: per-edge head softmax + q-weighted combine of T[src] rows,
// atomic scatter to acc[dst]. Edges [0,nE) from edge_index; [nE, nE+n) are
// implicit self-loops (PyG default). cnt accumulated only on the first conv
// (identical edge list for both convs).
// ---------------------------------------------------------------------------
template <int OUT>
__global__ void feast_edge(const int* __restrict__ ei,    // [2,nE] flat
                           const float* __restrict__ T,   // [n, 4*OUT]
                           const float* __restrict__ D,   // [n, 4]
                           const float* __restrict__ C,   // [4]
                           float* __restrict__ acc,       // [n, OUT]
                           float* __restrict__ cnt,       // [n] or nullptr
                           int nE, int n) {
    int e = blockIdx.x * blockDim.x + threadIdx.x;
    if (e >= nE + n) return;
    int src, dst;
    if (e < nE) { src = ei[e]; dst = ei[nE + e]; }
    else        { src = dst = e - nE; }

    float l[4];
#pragma unroll
    for (int h = 0; h < 4; ++h)
        l[h] = D[(size_t)src * 4 + h] - D[(size_t)dst * 4 + h] + C[h];
    float m = fmaxf(fmaxf(l[0], l[1]), fmaxf(l[2], l[3]));
    float q[4], s = 0.0f;
#pragma unroll
    for (int h = 0; h < 4; ++h) { q[h] = __expf(l[h] - m); s += q[h]; }
    float inv = 1.0f / s;
#pragma unroll
    for (int h = 0; h < 4; ++h) q[h] *= inv;

    const float* t = T + (size_t)src * (4 * OUT);
    float* arow = acc + (size_t)dst * OUT;
#pragma unroll
    for (int c0 = 0; c0 < OUT; c0 += 4) {
        float4 m4 = make_float4(0.f, 0.f, 0.f, 0.f);
#pragma unroll
        for (int h = 0; h < 4; ++h) {
            const float4 tv = *(const float4*)(t + h * OUT + c0);
            m4.x += q[h] * tv.x; m4.y += q[h] * tv.y;
            m4.z += q[h] * tv.z; m4.w += q[h] * tv.w;
        }
        atomicAdd(arow + c0 + 0, m4.x);
        atomicAdd(arow + c0 + 1, m4.y);
        atomicAdd(arow + c0 + 2, m4.z);
        atomicAdd(arow + c0 + 3, m4.w);
    }
    if (cnt) atomicAdd(&cnt[dst], 1.0f);
}

// ---------------------------------------------------------------------------
// In-place mean + bias + ReLU (conv1 epilogue)
// ---------------------------------------------------------------------------
__global__ void mean_bias_relu16(float* __restrict__ acc, const float* __restrict__ cnt,
                                 const float* __restrict__ b, int n) {
    int i = blockIdx.x * blockDim.x + threadIdx.x;
    if (i >= n * 16) return;
    int node = i >> 4, c = i & 15;
    float v = acc[i] / fmaxf(cnt[node], 1.0f) + b[c];
    acc[i] = fmaxf(v, 0.0f);
}

// ---------------------------------------------------------------------------
// Conv2 epilogue fused with the MLP stack 32->16->8->4->1 + sigmoid.
// Weights are tiny and L2/scalar-cache resident.
// ---------------------------------------------------------------------------
__global__ void final_mlp(const float* __restrict__ acc2, const float* __restrict__ cnt,
                          const float* __restrict__ b2,
                          const float* __restrict__ lw1, const float* __restrict__ lb1,
                          const float* __restrict__ lw2, const float* __restrict__ lb2,
                          const float* __restrict__ lw3, const float* __restrict__ lb3,
                          const float* __restrict__ low, const float* __restrict__ lob,
                          float* __restrict__ out, int n) {
    int i = blockIdx.x * blockDim.x + threadIdx.x;
    if (i >= n) return;
    float invc = 1.0f / fmaxf(cnt[i], 1.0f);
    float h0[32];
#pragma unroll
    for (int c = 0; c < 32; ++c)
        h0[c] = fmaxf(acc2[(size_t)i * 32 + c] * invc + b2[c], 0.0f);
    float h1[16];
#pragma unroll
    for (int o = 0; o < 16; ++o) {
        float s = lb1[o];
#pragma unroll
        for (int k = 0; k < 32; ++k) s += h0[k] * lw1[k * 16 + o];
        h1[o] = fmaxf(s, 0.0f);
    }
    float h2[8];
#pragma unroll
    for (int o = 0; o < 8; ++o) {
        float s = lb2[o];
#pragma unroll
        for (int k = 0; k < 16; ++k) s += h1[k] * lw2[k * 8 + o];
        h2[o] = fmaxf(s, 0.0f);
    }
    float h3[4];
#pragma unroll
    for (int o = 0; o < 4; ++o) {
        float s = lb3[o];
#pragma unroll
        for (int k = 0; k < 8; ++k) s += h2[k] * lw3[k * 4 + o];
        h3[o] = fmaxf(s, 0.0f);
    }
    float z = lob[0];
#pragma unroll
    for (int k = 0; k < 4; ++k) z += h3[k] * low[k];
    out[i] = 1.0f / (1.0f + __expf(-z));
}

// ---------------------------------------------------------------------------
extern "C" void kernel_launch(void* const* d_in, const int* in_sizes, int n_in,
                              void* d_out, int out_size, void* d_ws, size_t ws_size,
                              hipStream_t stream) {
    const float* x   = (const float*)d_in[0];
    const int*   ei  = (const int*)  d_in[1];
    const float* W1  = (const float*)d_in[2];
    const float* u1  = (const float*)d_in[3];
    const float* c1  = (const float*)d_in[4];
    const float* b1  = (const float*)d_in[5];
    const float* W2  = (const float*)d_in[6];
    const float* u2  = (const float*)d_in[7];
    const float* c2  = (const float*)d_in[8];
    const float* b2  = (const float*)d_in[9];
    const float* lw1 = (const float*)d_in[10];
    const float* lb1 = (const float*)d_in[11];
    const float* lw2 = (const float*)d_in[12];
    const float* lb2 = (const float*)d_in[13];
    const float* lw3 = (const float*)d_in[14];
    const float* lb3 = (const float*)d_in[15];
    const float* low = (const float*)d_in[16];
    const float* lob = (const float*)d_in[17];
    float* out = (float*)d_out;

    const int n  = N_NODES;
    const int nE = in_sizes[1] / 2;

    // Workspace layout (floats). T region reused for T1 (N*64) then T2 (N*128);
    // D region reused for d1 then d2. acc1 becomes h1 in place. ~72 MB total.
    float* ws   = (float*)d_ws;
    float* T    = ws;                         // N*128
    float* D    = T    + (size_t)n * 128;     // N*4
    float* acc1 = D    + (size_t)n * 4;       // N*16  (becomes h1)
    float* cnt  = acc1 + (size_t)n * 16;      // N
    float* acc2 = cnt  + (size_t)n;           // N*32

    // zero acc1|cnt|acc2 (contiguous N*49 floats)
    {
        size_t nz = (size_t)n * 49;
        zero_kernel<<<(unsigned)((nz + 255) / 256), 256, 0, stream>>>(acc1, nz);
    }

    const int tiles   = (n + 15) / 16;
    const int gblocks = (tiles + 7) / 8;          // 8 waves (256 thr) per block
    const int etot    = nE + n;
    const int eblocks = (etot + 255) / 256;

    // conv1
    feast_node_gemm<64><<<gblocks, 256, 0, stream>>>(x, W1, u1, T, D, n);
    feast_edge<16><<<eblocks, 256, 0, stream>>>(ei, T, D, c1, acc1, cnt, nE, n);
    mean_bias_relu16<<<(n * 16 + 255) / 256, 256, 0, stream>>>(acc1, cnt, b1, n);

    // conv2 (h1 lives in acc1)
    feast_node_gemm<128><<<gblocks, 256, 0, stream>>>(acc1, W2, u2, T, D, n);
    feast_edge<32><<<eblocks, 256, 0, stream>>>(ei, T, D, c2, acc2, nullptr, nE, n);

    // conv2 epilogue + MLP + sigmoid
    final_mlp<<<(n + 255) / 256, 256, 0, stream>>>(acc2, cnt, b2,
                                                   lw1, lb1, lw2, lb2, lw3, lb3,
                                                   low, lob, out, n);
}